// DenseGridEncoding_85727547228356
// MI455X (gfx1250) — compile-verified
//
#include <hip/hip_runtime.h>

// Dense grid trilinear encoding for MI455X (gfx1250).
//
// Memory-bound gather kernel. Key MI455X reasoning:
//  * grid = 128^3 x 20 f32 = 168 MB  -> fits in the 192 MB global L2 and stays
//    resident across graph replays; grid gathers use default RT cache policy.
//  * out  = 80 MB, x = 12 MB          -> streamed with NT cache policy
//    (global_load_b96/global_store_b128 th:*_NT) so they don't evict the grid.
//  * 8 corner rows x 80 B per point, each row 16B-aligned -> 40 independent
//    global_load_b128 gathers per thread for maximal memory-level parallelism.
//  * Weight divisions removed: cell width is exactly 2^-6 in f32, and IEEE
//    division by a power of two == multiplication by its reciprocal (exact),
//    so (x2-px)/dx == (x2-px)*64 bit-for-bit. Saves ~60 VALU ops/point.
//  * WMMA is deliberately NOT used: the 8x20 corner matrix differs per point
//    (B operand cannot be shared across D rows), and at ~1.6 flop/byte the
//    kernel is nowhere near any matrix-op roofline.

typedef __attribute__((ext_vector_type(4))) float v4f;

#define GRID_V   128
#define NUM_EMB  (GRID_V * GRID_V * GRID_V)   // 2097152
#define D_EMB    20                           // 5 float4 per row, 80 B, 16B aligned

__global__ __launch_bounds__(256) void dense_grid_enc_kernel(
    const float* __restrict__ x,
    const float* __restrict__ grid,
    float* __restrict__ out,
    int N)
{
    const int i = blockIdx.x * blockDim.x + threadIdx.x;
    if (i >= N) return;

    // x is streamed once per launch -> non-temporal loads keep L2 for the grid.
    const float px = __builtin_nontemporal_load(&x[3 * i + 0]);
    const float py = __builtin_nontemporal_load(&x[3 * i + 1]);
    const float pz = __builtin_nontemporal_load(&x[3 * i + 2]);

    // gp = floor((x - DOM_LO) / DOM_LEN * V), DOM = [-1, 1]
    const float fx = (px + 1.0f) * 0.5f * 128.0f;
    const float fy = (py + 1.0f) * 0.5f * 128.0f;
    const float fz = (pz + 1.0f) * 0.5f * 128.0f;
    const float gxf = floorf(fx), gyf = floorf(fy), gzf = floorf(fz);
    const int gx = (int)gxf, gy = (int)gyf, gz = (int)gzf;

    // Corner positions: pos(g) = g/64 - 1 (exact powers-of-two scaling, so
    // x1, x2 and dx = x2 - x1 = 2^-6 are all exact f32 values).
    const float x1 = gxf * (1.0f / 64.0f) - 1.0f;
    const float x2 = (gxf + 1.0f) * (1.0f / 64.0f) - 1.0f;
    const float y1 = gyf * (1.0f / 64.0f) - 1.0f;
    const float y2 = (gyf + 1.0f) * (1.0f / 64.0f) - 1.0f;
    const float z1 = gzf * (1.0f / 64.0f) - 1.0f;
    const float z2 = (gzf + 1.0f) * (1.0f / 64.0f) - 1.0f;

    // Trilinear weights: bit-identical to the reference's quotient form,
    // since dividing by the exact power-of-two dx is an exact scaling.
    const float wx0 = (x2 - px) * 64.0f, wx1 = (px - x1) * 64.0f;
    const float wy0 = (y2 - py) * 64.0f, wy1 = (py - y1) * 64.0f;
    const float wz0 = (z2 - pz) * 64.0f, wz1 = (pz - z1) * 64.0f;

    float w[8];
    w[0] = wx0 * wy0 * wz0;  // (0,0,0)
    w[1] = wx1 * wy0 * wz0;  // (1,0,0)
    w[2] = wx0 * wy1 * wz0;  // (0,1,0)
    w[3] = wx1 * wy1 * wz0;  // (1,1,0)
    w[4] = wx0 * wy0 * wz1;  // (0,0,1)
    w[5] = wx1 * wy0 * wz1;  // (1,0,1)
    w[6] = wx0 * wy1 * wz1;  // (0,1,1)
    w[7] = wx1 * wy1 * wz1;  // (1,1,1)

    // Flat indices; clamp replicates JAX clip-mode gather at the +1 boundary
    // (gp can reach 127 -> corner coordinate 128 -> flat >= NUM_EMB).
    const int base = gx + (gy << 7) + (gz << 14);
    int idx[8];
    idx[0] = base;
    idx[1] = base + 1;
    idx[2] = base + 128;
    idx[3] = base + 129;
    idx[4] = base + 16384;
    idx[5] = base + 16385;
    idx[6] = base + 16512;
    idx[7] = base + 16513;
#pragma unroll
    for (int c = 1; c < 8; ++c) {
        idx[c] = idx[c] < (NUM_EMB - 1) ? idx[c] : (NUM_EMB - 1);
    }

    // Gather-and-blend: 8 corners x 5 float4 (global_load_b128), default RT
    // policy so the grid stays resident in the 192 MB L2.
    v4f acc0 = 0.0f, acc1 = 0.0f, acc2 = 0.0f, acc3 = 0.0f, acc4 = 0.0f;
#pragma unroll
    for (int c = 0; c < 8; ++c) {
        const v4f* q = (const v4f*)(grid + (size_t)idx[c] * D_EMB);
        const v4f q0 = q[0];
        const v4f q1 = q[1];
        const v4f q2 = q[2];
        const v4f q3 = q[3];
        const v4f q4 = q[4];
        const float wc = w[c];
        acc0 += wc * q0;
        acc1 += wc * q1;
        acc2 += wc * q2;
        acc3 += wc * q3;
        acc4 += wc * q4;
    }

    // Output is 80 MB streamed once -> non-temporal stores protect L2.
    v4f* o = (v4f*)(out + (size_t)i * D_EMB);
    __builtin_nontemporal_store(acc0, &o[0]);
    __builtin_nontemporal_store(acc1, &o[1]);
    __builtin_nontemporal_store(acc2, &o[2]);
    __builtin_nontemporal_store(acc3, &o[3]);
    __builtin_nontemporal_store(acc4, &o[4]);
}

extern "C" void kernel_launch(void* const* d_in, const int* in_sizes, int n_in,
                              void* d_out, int out_size, void* d_ws, size_t ws_size,
                              hipStream_t stream) {
    const float* x    = (const float*)d_in[0];   // (N, 3) f32
    const float* grid = (const float*)d_in[1];   // (128^3, 20) f32
    float* out        = (float*)d_out;           // (N, 20) f32

    const int N = in_sizes[0] / 3;
    const int threads = 256;                      // 8 wave32 per block
    const int blocks  = (N + threads - 1) / threads;

    dense_grid_enc_kernel<<<blocks, threads, 0, stream>>>(x, grid, out, N);
}